// DynamicRoutingRNNCoarse_11622181503024
// MI455X (gfx1250) — compile-verified
//
#include <hip/hip_runtime.h>
#include <math.h>

typedef float v2f __attribute__((ext_vector_type(2)));
typedef float v8f __attribute__((ext_vector_type(8)));
typedef int   v4i __attribute__((vector_size(16)));

#define N_UNITS 100
#define NP      112      // padded units: 7 tiles of 16
#define ROWP    101      // LDS row stride for W / ym (coprime with 64 banks; cols 0..99 only are read)
#define T_STEPS 60
#define NTILES  7
#define N_IN    7

typedef __attribute__((address_space(1))) v4i* glb_v4p;   // global-memory v4i*
typedef __attribute__((address_space(3))) v4i* lds_v4p;   // LDS v4i*

__device__ __forceinline__ void async_copy_b128(const float* g, float* l) {
#if __has_builtin(__builtin_amdgcn_global_load_async_to_lds_b128)
    __builtin_amdgcn_global_load_async_to_lds_b128(
        (glb_v4p)(void*)(const_cast<float*>(g)), (lds_v4p)l, 0, 0);
#else
    lds_v4p l3 = (lds_v4p)l;
    glb_v4p g1 = (glb_v4p)(void*)(const_cast<float*>(g));
    asm volatile("global_load_async_to_lds_b128 %0, %1, off" : : "v"(l3), "v"(g1) : "memory");
#endif
}

__device__ __forceinline__ void wait_async0() {
#if __has_builtin(__builtin_amdgcn_s_wait_asynccnt)
    __builtin_amdgcn_s_wait_asynccnt(0);
#else
    asm volatile("s_wait_asynccnt 0x0" : : : "memory");
#endif
}

__global__ __launch_bounds__(32)
void dynroute_rnn_kernel(const float* __restrict__ y0,
                         const float* __restrict__ noise,
                         const int*   __restrict__ stimulus_idx,
                         const int*   __restrict__ rewarded_idx,
                         const int*   __restrict__ instruction,
                         const float* __restrict__ W_in_raw,
                         const float* __restrict__ W_rec,
                         const float* __restrict__ b_rec,
                         const float* __restrict__ w_out,
                         const float* __restrict__ b_out,
                         float* __restrict__ out,
                         int B)
{
    __shared__ float Wlds[NP * ROWP];            // W_rec padded: row j (output unit), col k (0..99)
    __shared__ float Win[N_IN * NP];             // |W_in| by channel, padded with zeros
    __shared__ float wout[NP];
    __shared__ float brec[NP];
    __shared__ float ym[16 * ROWP];              // y^T staging: row = trial lane, col = unit
    __shared__ __align__(16) float nbuf[16 * N_UNITS]; // async-staged noise slab for current t

    const int lane  = threadIdx.x;               // 0..31
    const int tl    = lane & 15;                 // trial within tile
    const int half  = lane >> 4;                 // 0 or 1
    const int tb    = blockIdx.x * 16;
    const int trial = tb + tl;

    const float kALPHA = 0.2f;
    const float kNOISE = 0.09486832980505137f;   // 0.15*sqrt(2*0.2)

    // ---- load shared weights (padded region zeroed) ----
    for (int i = lane; i < NP * ROWP; i += 32) {
        int j = i / ROWP, k = i - j * ROWP;
        Wlds[i] = (j < N_UNITS && k < N_UNITS) ? W_rec[j * N_UNITS + k] : 0.f;
    }
    for (int i = lane; i < N_IN * NP; i += 32) {
        int c = i / NP, j = i - c * NP;
        Win[i] = (j < N_UNITS) ? fabsf(W_in_raw[j * N_IN + c]) : 0.f;
    }
    for (int i = lane; i < NP; i += 32) {
        wout[i] = (i < N_UNITS) ? w_out[i] : 0.f;
        brec[i] = (i < N_UNITS) ? b_rec[i] : 0.f;
    }
    __syncthreads();

    // ---- per-trial constants (lane-local, replicated in both halves) ----
    const int  stim    = stimulus_idx[trial];
    const bool is_rew  = (stim == rewarded_idx[trial]);
    const bool instr_b = (instruction[trial] > 0);
    const float b_out0 = b_out[0];

    // y state in C/D fragment layout: unit j = jt*16 + v + 8*half, trial = tl
    float yreg[NTILES][8];
#pragma unroll
    for (int jt = 0; jt < NTILES; ++jt)
#pragma unroll
        for (int v = 0; v < 8; ++v) {
            int j = jt * 16 + v + 8 * half;
            yreg[jt][v] = (j < N_UNITS) ? y0[(size_t)trial * N_UNITS + j] : 0.f;
        }

    bool licked = false, instr_fired = false;
    int  lick_t = T_STEPS + 1;

    // output layout (tuple order, flattened)
    float* uo  = out;
    float* tgt = uo  + (size_t)B * T_STEPS * N_IN;
    float* rsp = tgt + (size_t)B * T_STEPS;
    float* lck = rsp + (size_t)B * T_STEPS;
    float* rwd = lck + (size_t)B;
    float* zq  = rwd + (size_t)B;
    float* yf  = zq  + (size_t)B * T_STEPS;

    for (int t = 0; t < T_STEPS; ++t) {
        const float stim_mask = (t >= 10 && t < 15) ? 1.f : 0.f;
        const float resp_mask = (t >= 20 && t < 35) ? 1.f : 0.f;

        // ---- kick off async HBM->LDS copy of this step's noise slab (16 trials x 100 f32)
        //      13 x B128 lane-copies; latency hidden behind the WMMA phase below.
#pragma unroll
        for (int q = 0; q < 13; ++q) {
            int e = q * 32 + lane;               // 16B chunk id, 400 chunks total
            if (e < 400) {
                int tr = e / 25, ch = e - tr * 25;
                const float* g = noise + ((size_t)(tb + tr) * T_STEPS + t) * N_UNITS + ch * 4;
                async_copy_b128(g, nbuf + e * 4);
            }
        }

        // ---- warm L2 one step ahead (4 x 128B lines per trial row, 2 per lane) ----
        if (t + 1 < T_STEPS) {
            int tr2 = lane >> 1;
            const float* nrow = noise + ((size_t)(tb + tr2) * T_STEPS + (t + 1)) * N_UNITS;
            __builtin_prefetch(nrow + (lane & 1) * 32, 0, 0);
            __builtin_prefetch(nrow + 64 + (lane & 1) * 32, 0, 0);
        }

        // ---- control pre-phase (matches reference step() ordering) ----
        bool delivered    = instr_fired || (licked && is_rew);
        bool fire         = instr_b && !delivered && (t == 30);
        instr_fired       = instr_fired || fire;
        bool instr_active = instr_fired && (t >= 30) && (t < 35);
        bool lick_dyn     = licked && (t > lick_t) && (t < lick_t + 5);
        float cR = (instr_active ? 1.f : 0.f) + ((lick_dyn && is_rew) ? 1.f : 0.f);
        float cL = lick_dyn ? 1.f : 0.f;

        // ---- publish y to LDS as B-matrix source (trial-major rows) ----
#pragma unroll
        for (int jt = 0; jt < NTILES; ++jt)
#pragma unroll
            for (int v = 0; v < 8; ++v) {
                int j = jt * 16 + v + 8 * half;
                ym[tl * ROWP + j] = yreg[jt][v];
            }
        __syncthreads();   // single-wave WG: s_wait_dscnt + s_nop

        // ---- pre = W_rec x y^T via fp32 WMMA, D[j][trial] ----
        v8f acc[NTILES];
#pragma unroll
        for (int jt = 0; jt < NTILES; ++jt)
#pragma unroll
            for (int e = 0; e < 8; ++e) acc[jt][e] = 0.f;

        for (int k0 = 0; k0 < N_UNITS; k0 += 4) {   // 25 K-tiles, K=100 exact
            v2f bfrag;
            int bbase = tl * ROWP + k0 + 2 * half;  // B: k = k0 + v + 2*half, n = tl
            bfrag.x = ym[bbase];
            bfrag.y = ym[bbase + 1];
#pragma unroll
            for (int jt = 0; jt < NTILES; ++jt) {
                v2f afrag;
                int abase = (jt * 16 + tl) * ROWP + k0 + 2 * half; // A: m = tl, k = k0 + v + 2*half
                afrag.x = Wlds[abase];
                afrag.y = Wlds[abase + 1];
                acc[jt] = __builtin_amdgcn_wmma_f32_16x16x4_f32(
                    false, afrag, false, bfrag, (short)0, acc[jt], false, false);
            }
        }
        __syncthreads();   // all fragment reads done before next-iter ym stores

        // ---- noise slab has landed in LDS ----
        wait_async0();

        // ---- elementwise: drive + injections + noise, leaky ReLU update ----
#pragma unroll
        for (int jt = 0; jt < NTILES; ++jt)
#pragma unroll
            for (int v = 0; v < 8; ++v) {
                int j = jt * 16 + v + 8 * half;
                float wu  = resp_mask * Win[6 * NP + j]
                          + stim_mask * Win[stim * NP + j] + brec[j];
                float inj = cR * Win[4 * NP + j] + cL * Win[5 * NP + j];
                float eps = (j < N_UNITS) ? nbuf[tl * N_UNITS + j] : 0.f;
                float pre = acc[jt][v] + wu + inj + kNOISE * eps;
                yreg[jt][v] = (1.f - kALPHA) * yreg[jt][v] + kALPHA * fmaxf(pre, 0.f);
            }

        // ---- readout: z = sigmoid(y . w_out + b_out), lane-half reduction ----
        float p = 0.f;
#pragma unroll
        for (int jt = 0; jt < NTILES; ++jt)
#pragma unroll
            for (int v = 0; v < 8; ++v) {
                int j = jt * 16 + v + 8 * half;
                p += yreg[jt][v] * wout[j];
            }
        p += __shfl_xor(p, 16, 32);
        float z = 1.f / (1.f + expf(-(p + b_out0)));

        // ---- control post-phase ----
        bool in_resp = (t >= 20) && (t < 35);
        bool trig    = in_resp && !licked && (z > 0.5f);
        lick_t       = trig ? t : lick_t;
        licked       = licked || trig;
        bool u5      = licked && (t >= lick_t) && (t < lick_t + 5);
        bool u4      = instr_active || (u5 && is_rew);

        // ---- per-timestep outputs (one half writes) ----
        if (half == 0) {
            size_t rowt = (size_t)trial * T_STEPS + t;
            float* up = uo + rowt * N_IN;
            up[0] = (stim == 0) ? stim_mask : 0.f;
            up[1] = (stim == 1) ? stim_mask : 0.f;
            up[2] = (stim == 2) ? stim_mask : 0.f;
            up[3] = (stim == 3) ? stim_mask : 0.f;
            up[4] = u4 ? 1.f : 0.f;
            up[5] = u5 ? 1.f : 0.f;
            up[6] = resp_mask;
            tgt[rowt] = is_rew ? resp_mask : 0.f;
            rsp[rowt] = resp_mask;
            zq[rowt]  = z;
        }
    }

    // ---- final outputs ----
    if (half == 0) {
        lck[trial] = licked ? 1.f : 0.f;
        bool rd = instr_fired || (licked && is_rew);
        rwd[trial] = rd ? 1.f : 0.f;
    }
#pragma unroll
    for (int jt = 0; jt < NTILES; ++jt)
#pragma unroll
        for (int v = 0; v < 8; ++v) {
            int j = jt * 16 + v + 8 * half;
            if (j < N_UNITS) yf[(size_t)trial * N_UNITS + j] = yreg[jt][v];
        }
}

extern "C" void kernel_launch(void* const* d_in, const int* in_sizes, int n_in,
                              void* d_out, int out_size, void* d_ws, size_t ws_size,
                              hipStream_t stream) {
    const float* y0       = (const float*)d_in[0];
    const float* noise    = (const float*)d_in[1];
    const int*   stim     = (const int*)  d_in[2];
    const int*   rew      = (const int*)  d_in[3];
    const int*   instr    = (const int*)  d_in[4];
    const float* W_in_raw = (const float*)d_in[5];
    const float* W_rec    = (const float*)d_in[6];
    const float* b_rec    = (const float*)d_in[7];
    const float* w_out    = (const float*)d_in[8];
    const float* b_out    = (const float*)d_in[9];
    float* out = (float*)d_out;

    int B = in_sizes[2];            // stimulus_idx count = batch
    dim3 grid((B + 15) / 16), block(32);
    dynroute_rnn_kernel<<<grid, block, 0, stream>>>(
        y0, noise, stim, rew, instr, W_in_raw, W_rec, b_rec, w_out, b_out, out, B);
}